// NodeDropout_32770600468631
// MI455X (gfx1250) — compile-verified
//
#include <hip/hip_runtime.h>
#include <stdint.h>

// ---------------------------------------------------------------------------
// NodeDropout edge compaction for MI455X (gfx1250, wave32).
//   * memory-bound stream compaction: ~640MB traffic -> ~27us @ 23.3 TB/s
//   * 25KB drop-flag bitmask DMA'd into LDS via TDM (tensor_load_to_lds)
//   * wave32 ballot + LDS scan for stable intra-block ranks
// ---------------------------------------------------------------------------

#define TPB         256
#define NBLK        1024
#define NUM_USERS   100000
#define NUM_ITEMS   100000
#define TOTAL_NODES (NUM_USERS + NUM_ITEMS)
#define FLAG_WORDS  (TOTAL_NODES / 32)     /* 6250 u32 words = 25000 B */
#define DROP_THRESH 429496729u             /* ~0.1 * 2^32 */

typedef unsigned int uint;
typedef unsigned int u32x4 __attribute__((ext_vector_type(4)));
typedef int          i32x8 __attribute__((ext_vector_type(8)));
typedef int          i32x4 __attribute__((ext_vector_type(4)));

#if defined(__has_builtin)
#  if __has_builtin(__builtin_amdgcn_tensor_load_to_lds)
#    define HAVE_TDM 1
#  endif
#  if __has_builtin(__builtin_amdgcn_ballot_w32)
#    define HAVE_BALLOT32 1
#  endif
#endif

__device__ __forceinline__ uint hash32(uint x) {
    x ^= x >> 16; x *= 0x7feb352du;
    x ^= x >> 15; x *= 0x846ca68bu;
    x ^= x >> 16;
    return x;
}

// DMA the 25000-byte drop-flag table into LDS with the Tensor Data Mover.
// One wave issues the descriptor (TDM ignores EXEC; issues once per wave),
// waits on TENSORcnt, then the workgroup barrier publishes the LDS data.
__device__ __forceinline__ void load_flags(uint* sFlags, const uint* gFlags) {
#ifdef HAVE_TDM
    if (threadIdx.x < 32) {
        uint32_t lds = (uint32_t)(uintptr_t)sFlags;   // low 32 bits == LDS byte addr
        uint64_t ga  = (uint64_t)(uintptr_t)gFlags;

        u32x4 g0;
        g0.x = 1u;                                     // count=1, user descriptor
        g0.y = lds;                                    // lds_addr
        g0.z = (uint)(ga & 0xFFFFFFFFu);               // global_addr[31:0]
        g0.w = (uint)((ga >> 32) & 0x01FFFFFFu)        // global_addr[56:32]
             | (2u << 30);                             // type = 2 ("image")

        i32x8 g1;
        g1[0] = (int)(2u << 16);                       // workgroup_mask=0, data_size=4B
        g1[1] = (int)((uint)FLAG_WORDS << 16);         // tensor_dim0[15:0] in bits 63:48
        g1[2] = (int)(1u << 16);                       // tensor_dim0 hi=0, tensor_dim1=1
        g1[3] = (int)((uint)FLAG_WORDS << 16);         // tensor_dim1 hi=0, tile_dim0=6250
        g1[4] = 1;                                     // tile_dim1=1, tile_dim2=0
        g1[5] = FLAG_WORDS;                            // tensor_dim0_stride[31:0]
        g1[6] = 0;                                     // stride0 hi, stride1 lo
        g1[7] = 0;                                     // stride1 hi

        i32x4 g2; g2[0] = 1; g2[1] = 1; g2[2] = 0; g2[3] = 0;  // dim2=1, dim3=1
        i32x4 g3; g3[0] = 0; g3[1] = 0; g3[2] = 0; g3[3] = 0;
        i32x8 g4; g4[0] = 0; g4[1] = 0; g4[2] = 0; g4[3] = 0;  // extra group (6-arg
        g4[4] = 0; g4[5] = 0; g4[6] = 0; g4[7] = 0;            //  clang-23 form)

        __builtin_amdgcn_tensor_load_to_lds(g0, g1, g2, g3, g4, 0);
#  if __has_builtin(__builtin_amdgcn_s_wait_tensorcnt)
        __builtin_amdgcn_s_wait_tensorcnt(0);
#  else
        asm volatile("s_wait_tensorcnt 0x0" ::: "memory");
#  endif
    }
    __syncthreads();
#else
    for (int i = threadIdx.x; i < FLAG_WORDS; i += blockDim.x) sFlags[i] = gFlags[i];
    __syncthreads();
#endif
}

__device__ __forceinline__ bool edge_kept(const uint* sFlags, int u, int v) {
    uint du = (sFlags[((uint)u) >> 5] >> (u & 31)) & 1u;
    uint dv = (sFlags[((uint)v) >> 5] >> (v & 31)) & 1u;
    return (du & dv) == 0u;    // keep if at least one endpoint survives
}

__device__ __forceinline__ uint ballot32(bool p) {
#ifdef HAVE_BALLOT32
    return __builtin_amdgcn_ballot_w32(p);
#else
    return (uint)__ballot(p);  // wave32: low 32 bits valid
#endif
}

// --- K0: deterministic drop-flag bitmask (one u32 word = 32 nodes) ---------
__global__ void dropflag_kernel(uint* __restrict__ flags) {
    int w = blockIdx.x * blockDim.x + threadIdx.x;
    if (w >= FLAG_WORDS) return;
    uint word = 0u;
    for (int b = 0; b < 32; ++b) {
        uint node = (uint)w * 32u + (uint)b;
        if (hash32(node * 0x9E3779B9u + 42u) < DROP_THRESH) word |= (1u << b);
    }
    flags[w] = word;
}

// --- K1: per-block kept-edge counts ----------------------------------------
__global__ void count_kernel(const int* __restrict__ ei, const uint* __restrict__ gFlags,
                             uint* __restrict__ counts, int E, int epb) {
    __shared__ uint sFlags[FLAG_WORDS];
    __shared__ uint sRed[TPB];
    load_flags(sFlags, gFlags);

    int b = blockIdx.x;
    int chunkStart = b * epb;
    int chunkEnd   = min(E, chunkStart + epb);

    uint cnt = 0;
    for (int i = chunkStart + (int)threadIdx.x; i < chunkEnd; i += TPB) {
        __builtin_prefetch(&ei[i + 8 * TPB], 0, 0);        // global_prefetch_b8
        __builtin_prefetch(&ei[E + i + 8 * TPB], 0, 0);
        int u = ei[i], v = ei[E + i];
        cnt += edge_kept(sFlags, u, v) ? 1u : 0u;
    }
    sRed[threadIdx.x] = cnt;
    __syncthreads();
    for (int s = TPB / 2; s > 0; s >>= 1) {
        if ((int)threadIdx.x < s) sRed[threadIdx.x] += sRed[threadIdx.x + s];
        __syncthreads();
    }
    if (threadIdx.x == 0) counts[b] = sRed[0];
}

// --- K2: exclusive scan of NBLK block counts (single block) ----------------
__global__ void scan_kernel(const uint* __restrict__ counts, uint* __restrict__ offs) {
    __shared__ uint s[NBLK];
    int tid = threadIdx.x;
    uint mine = counts[tid];
    s[tid] = mine;
    for (int off = 1; off < NBLK; off <<= 1) {
        __syncthreads();
        uint t = (tid >= off) ? s[tid - off] : 0u;
        __syncthreads();
        s[tid] += t;
    }
    offs[tid] = s[tid] - mine;                 // exclusive offset
    if (tid == NBLK - 1) offs[NBLK] = s[tid];  // total kept = n_keep
}

// --- K3: stable scatter: kept edges to front, -1 tail ----------------------
__global__ void scatter_kernel(const int* __restrict__ ei, int* __restrict__ out,
                               const uint* __restrict__ gFlags, const uint* __restrict__ offs,
                               int E, int epb) {
    __shared__ uint sFlags[FLAG_WORDS];
    __shared__ uint sWave[TPB / 32];
    __shared__ uint sTileTotal;
    load_flags(sFlags, gFlags);

    int b = blockIdx.x;
    int chunkStart = b * epb;
    int chunkEnd   = min(E, chunkStart + epb);
    if (chunkStart >= chunkEnd) return;        // uniform exit, no barriers after

    uint keepBase = offs[b];
    uint nKeep    = offs[NBLK];
    uint dropBase = nKeep + (uint)chunkStart - keepBase;  // n_keep + dropped before block

    int  tid  = (int)threadIdx.x;
    int  lane = tid & 31;
    int  wid  = tid >> 5;
    uint keptSoFar = 0, processed = 0;

    for (int base = chunkStart; base < chunkEnd; base += TPB) {
        int  idx   = base + tid;
        bool valid = idx < chunkEnd;
        int  u = 0, v = 0;
        bool keep = false;
        if (valid) {
            __builtin_prefetch(&ei[idx + 8 * TPB], 0, 0);
            __builtin_prefetch(&ei[E + idx + 8 * TPB], 0, 0);
            u = ei[idx]; v = ei[E + idx];
            keep = edge_kept(sFlags, u, v);
        }
        uint mask    = ballot32(keep);
        uint wPrefix = __popc(mask & ((1u << lane) - 1u));
        if (lane == 0) sWave[wid] = (uint)__popc(mask);
        __syncthreads();
        if (tid == 0) {
            uint r = 0;
            for (int w = 0; w < TPB / 32; ++w) { uint t = sWave[w]; sWave[w] = r; r += t; }
            sTileTotal = r;
        }
        __syncthreads();
        uint localPrefix = sWave[wid] + wPrefix;  // kept before me inside tile
        uint tileTotal   = sTileTotal;
        if (valid) {
            if (keep) {
                uint p = keepBase + keptSoFar + localPrefix;
                out[p]     = u;
                out[E + p] = v;
            } else {
                uint dprefix = processed + (uint)tid - (keptSoFar + localPrefix);
                uint p = dropBase + dprefix;
                out[p]     = -1;
                out[E + p] = -1;
            }
        }
        keptSoFar += tileTotal;
        processed += (uint)min(TPB, chunkEnd - base);
        __syncthreads();                       // sWave/sTileTotal reuse next tile
    }
}

// ---------------------------------------------------------------------------
extern "C" void kernel_launch(void* const* d_in, const int* in_sizes, int n_in,
                              void* d_out, int out_size, void* d_ws, size_t ws_size,
                              hipStream_t stream) {
    (void)n_in; (void)out_size; (void)ws_size;

    const int* ei  = (const int*)d_in[0];
    int        E   = in_sizes[0] / 2;
    int*       out = (int*)d_out;

    char* ws     = (char*)d_ws;
    uint* flags  = (uint*)(ws);            // 25000 B bitmask
    uint* counts = (uint*)(ws + 25600);    // NBLK u32
    uint* offs   = (uint*)(ws + 32768);    // NBLK+1 u32

    int epb = (E + NBLK - 1) / NBLK;

    dropflag_kernel<<<(FLAG_WORDS + TPB - 1) / TPB, TPB, 0, stream>>>(flags);
    count_kernel  <<<NBLK, TPB, 0, stream>>>(ei, flags, counts, E, epb);
    scan_kernel   <<<1, NBLK, 0, stream>>>(counts, offs);
    scatter_kernel<<<NBLK, TPB, 0, stream>>>(ei, out, flags, offs, E, epb);
}